// MaskFirst_20349555048660
// MI455X (gfx1250) — compile-verified
//
#include <hip/hip_runtime.h>
#include <math.h>

typedef __attribute__((ext_vector_type(2))) float v2f;
typedef __attribute__((ext_vector_type(8))) float v8f;

#define CCH 256   // feature channels
#define HW  256   // pixels per image (16x16)
#define TAPS 49   // 7x7 filter taps

// ---------------------------------------------------------------------------
// Kernel 1: Q[n, t, pix] = sum_c w[c*49 + t] * feat[n, c, pix]
// GEMM M=49(->64) x N=256 x K=256 in f32 via V_WMMA_F32_16X16X4_F32.
// One wave handles one M-tile x two N-tiles (two accumulators, shared A):
// grid = 2(n) * 4(mt) * 8(nt-pair) = 64 blocks of 32 threads.
// A loads are branchless (always in-bounds; zero-pad rows masked by multiply),
// fragment loads are software-pipelined one K-step ahead of the WMMAs.
// ---------------------------------------------------------------------------
__global__ __launch_bounds__(32) void gemm_q_kernel(const float* __restrict__ feat,
                                                    const float* __restrict__ wgt,
                                                    float* __restrict__ Q) {
  const int bid  = blockIdx.x;
  const int n    = bid >> 5;
  const int mt   = (bid >> 3) & 3;
  const int ntp  = bid & 7;          // pair of N-tiles
  const int lane = threadIdx.x & 31;
  const int half = lane >> 4;        // 0: K pair {0,1}, 1: K pair {2,3}
  const int l    = lane & 15;

  const int   row   = mt * 16 + l;                 // tap index (A row, M)
  const float amask = (row < TAPS) ? 1.f : 0.f;    // zero-pad rows 49..63
  const int   col0  = ntp * 32 + l;                // pixel (B col, tile 0)
  const int   col1  = col0 + 16;                   // pixel (B col, tile 1)
  const float* F = feat + n * CCH * HW;

  v8f acc0 = {0.f, 0.f, 0.f, 0.f, 0.f, 0.f, 0.f, 0.f};
  v8f acc1 = acc0;

  // Fragment loader for K-step k0 (this lane covers K = k0+2*half, +1).
  // Max wgt index: 255*49 + 63 = 12558 < 12593 -> unconditional load is safe.
  auto load_frag = [&](int k0, v2f& a, v2f& b0, v2f& b1) {
    const int ka = k0 + 2 * half;
    a.x  = wgt[ka * TAPS + row] * amask;
    a.y  = wgt[(ka + 1) * TAPS + row] * amask;
    b0.x = F[ka * HW + col0];
    b0.y = F[(ka + 1) * HW + col0];
    b1.x = F[ka * HW + col1];
    b1.y = F[(ka + 1) * HW + col1];
  };

  v2f a_c, b0_c, b1_c;
  load_frag(0, a_c, b0_c, b1_c);
  __builtin_prefetch(F + 16 * HW + col0, 0, 1);  // warm L2 ahead of the K walk

  for (int k0 = 0; k0 < CCH - 4; k0 += 4) {
    v2f a_n, b0_n, b1_n;
    load_frag(k0 + 4, a_n, b0_n, b1_n);  // in flight while WMMAs execute
    acc0 = __builtin_amdgcn_wmma_f32_16x16x4_f32(false, a_c, false, b0_c,
                                                 (short)0, acc0, false, false);
    acc1 = __builtin_amdgcn_wmma_f32_16x16x4_f32(false, a_c, false, b1_c,
                                                 (short)0, acc1, false, false);
    a_c = a_n; b0_c = b0_n; b1_c = b1_n;
  }
  acc0 = __builtin_amdgcn_wmma_f32_16x16x4_f32(false, a_c, false, b0_c,
                                               (short)0, acc0, false, false);
  acc1 = __builtin_amdgcn_wmma_f32_16x16x4_f32(false, a_c, false, b1_c,
                                               (short)0, acc1, false, false);

  // D layout: VGPR j -> row (j + 8*half), col l of each 16x16 tile
  float* Qn = Q + n * 64 * HW;
#pragma unroll
  for (int j = 0; j < 8; ++j) {
    const int trow = mt * 16 + j + 8 * half;
    Qn[trow * HW + col0] = acc0[j];
    Qn[trow * HW + col1] = acc1[j];
  }
}

// ---------------------------------------------------------------------------
// Kernel 2: per-image prep.
//  base[n,hw] = sum over valid taps of Q + b
//  sumt[n,hw] = #targets on at pixel (== cnt)
//  idx[n,p]   = argmin-area target on at p (first-min tie-break), -1 if none
// grid = 2 blocks (one per n), 256 threads (one per pixel).
// ---------------------------------------------------------------------------
__global__ __launch_bounds__(256) void prep_kernel(const float* __restrict__ bptr,
                                                   const int* __restrict__ targets,
                                                   const float* __restrict__ Q,
                                                   float* __restrict__ base,
                                                   float* __restrict__ sumt,
                                                   int* __restrict__ idxbuf) {
  const int n = blockIdx.x;
  const int j = threadIdx.x;
  const int h = j >> 4, x0 = j & 15;

  // base assembly: gather-add 49 shifted Q planes
  const float* Qn = Q + n * 64 * HW;
  float s = 0.f;
#pragma unroll
  for (int kh = 0; kh < 7; ++kh) {
    const int y = h + kh - 3;
    if ((unsigned)y < 16u) {
#pragma unroll
      for (int kw = 0; kw < 7; ++kw) {
        const int x = x0 + kw - 3;
        if ((unsigned)x < 16u) s += Qn[(kh * 7 + kw) * HW + y * 16 + x];
      }
    }
  }
  base[n * HW + j] = s + bptr[0];

  // target stats
  __shared__ int s_areas[8];
  if (j < 8) s_areas[j] = 0;
  __syncthreads();

  int tv[8];
  int cnt = 0;
#pragma unroll
  for (int t = 0; t < 8; ++t) {
    tv[t] = targets[(n * 8 + t) * HW + j];
    cnt += tv[t];
    if (tv[t]) atomicAdd(&s_areas[t], 1);  // integer LDS atomic: deterministic
  }
  __syncthreads();

  sumt[n * HW + j] = (float)cnt;

  float best = 3.4e38f;
  int bi = -1;
#pragma unroll
  for (int t = 0; t < 8; ++t) {  // ascending t + strict '<' == first-min argmin
    const float a = (float)s_areas[t];
    if (tv[t] && a < best) { best = a; bi = t; }
  }
  idxbuf[n * HW + j] = bi;  // -1 => cnt==0 => use loss_neg path
}

// ---------------------------------------------------------------------------
// Kernel 3: per-(n,p) loss partial. grid = 512 blocks, 256 threads (pixel j).
//  m = sigmoid(base[hw] + w_last[ph-h+3, pw-x+3])
//  contrib = |m - chosen| (cnt>0)  or  sumt*m (cnt==0); partial = mean over hw
// ---------------------------------------------------------------------------
__global__ __launch_bounds__(256) void loss_kernel(const float* __restrict__ wgt,
                                                   const int* __restrict__ targets,
                                                   const float* __restrict__ base,
                                                   const float* __restrict__ sumt,
                                                   const int* __restrict__ idxbuf,
                                                   float* __restrict__ part) {
  const int bid = blockIdx.x;          // n*256 + p
  const int n = bid >> 8, p = bid & 255;
  const int ph = p >> 4, pw = p & 15;
  const int j = threadIdx.x;
  const int h = j >> 4, x0 = j & 15;

  float xv = base[n * HW + j];
  const int kh = ph - h + 3, kw = pw - x0 + 3;
  if ((unsigned)kh < 7u && (unsigned)kw < 7u) xv += wgt[CCH * TAPS + kh * 7 + kw];
  const float m = 1.f / (1.f + expf(-xv));

  const int id = idxbuf[n * HW + p];   // uniform across block
  float contrib;
  if (id >= 0) {
    const float ch = (float)targets[(n * 8 + id) * HW + j];
    contrib = fabsf(m - ch);
  } else {
    contrib = sumt[n * HW + j] * m;
  }

  __shared__ float red[256];
  red[j] = contrib;
  __syncthreads();
  for (int s2 = 128; s2 > 0; s2 >>= 1) {
    if (j < s2) red[j] += red[j + s2];
    __syncthreads();
  }
  if (j == 0) part[bid] = red[0] * (1.f / 256.f);
}

// ---------------------------------------------------------------------------
// Kernel 4: deterministic fixed-tree reduce of 512 partials -> scalar loss.
// ---------------------------------------------------------------------------
__global__ __launch_bounds__(512) void reduce_kernel(const float* __restrict__ part,
                                                     float* __restrict__ out) {
  __shared__ float red[512];
  const int j = threadIdx.x;
  red[j] = part[j];
  __syncthreads();
  for (int s2 = 256; s2 > 0; s2 >>= 1) {
    if (j < s2) red[j] += red[j + s2];
    __syncthreads();
  }
  if (j == 0) out[0] = red[0];
}

extern "C" void kernel_launch(void* const* d_in, const int* in_sizes, int n_in,
                              void* d_out, int out_size, void* d_ws, size_t ws_size,
                              hipStream_t stream) {
  const float* feat    = (const float*)d_in[0];  // (2,256,16,16)
  const float* wgt     = (const float*)d_in[1];  // (1,257,7,7) flat: c*49 + kh*7 + kw
  const float* bptr    = (const float*)d_in[2];  // (1,)
  const int*   targets = (const int*)d_in[3];    // (2,8,16,16)

  // workspace layout (floats): Q[2*64*256] | base[512] | sumt[512] | idx[512] | part[512]
  float* Q    = (float*)d_ws;
  float* base = Q + 2 * 64 * HW;
  float* sumt = base + 2 * HW;
  int*   idxb = (int*)(sumt + 2 * HW);
  float* part = (float*)(idxb + 2 * HW);

  gemm_q_kernel<<<64, 32, 0, stream>>>(feat, wgt, Q);
  prep_kernel<<<2, 256, 0, stream>>>(bptr, targets, Q, base, sumt, idxb);
  loss_kernel<<<512, 256, 0, stream>>>(wgt, targets, base, sumt, idxb, part);
  reduce_kernel<<<1, 512, 0, stream>>>(part, (float*)d_out);
}